// VIN_30382598652390
// MI455X (gfx1250) — compile-verified
//
#include <hip/hip_runtime.h>

// VIN on MI455X (gfx1250, wave32).
// - Hidden 256-ch layer algebraically folded into a 2x3x3 kernel (prep kernel).
// - One workgroup (4 waves / 128 threads) per batch item; v in padded LDS tile,
//   qr kept in 128 VGPRs/lane as WMMA C-fragments.
// - Value-iteration conv done with V_WMMA_F32_16X16X4_F32 (f32-exact):
//   Q(64x256) = W(64x9) x P(9x256) + QR ; 4 M-tiles x 3 K-steps x 16 N-tiles,
//   each N-tile = one 16-pixel image row.
// - Hot loop uses precomputed loop-invariant LDS offsets + a zero slot
//   (vpad[324]) so all B-fragment loads are unconditional ds_load_b32,
//   and a raw ds_bpermute for the cross-half channel max.

#define KITER 40

typedef __attribute__((ext_vector_type(2))) float v2f;
typedef __attribute__((ext_vector_type(8))) float v8f;

__device__ __forceinline__ v8f wmma4(v2f a, v2f b, v8f c) {
  // 8 args: (neg_a, A, neg_b, B, c_mod, C, reuse_a, reuse_b)
  return __builtin_amdgcn_wmma_f32_16x16x4_f32(false, a, false, b, (short)0, c,
                                               false, false);
}

// A-matrix 16x4 f32 layout (wave32): lane = (k>=2 ? 16 : 0) + m ; vgpr = k & 1.
__device__ __forceinline__ void load_afrag(const float* __restrict__ W, int hi,
                                           int n, v2f af[4][3]) {
#pragma unroll
  for (int mt = 0; mt < 4; ++mt)
#pragma unroll
    for (int s = 0; s < 3; ++s)
#pragma unroll
      for (int r = 0; r < 2; ++r) {
        int k = 4 * s + 2 * hi + r;  // K index within 0..11 (9..11 = zero pad)
        af[mt][s][r] = (k < 9) ? W[(16 * mt + n) * 9 + k] : 0.f;
      }
}

// Fold r = conv1x1(conv3x3(X,h_w)+h_b, r_w) into one 2x3x3 kernel + bias.
__global__ void vin_prep_kernel(const float* __restrict__ h_w,
                                const float* __restrict__ h_b,
                                const float* __restrict__ r_w,
                                float* __restrict__ webe) {
  int j = threadIdx.x;
  if (j < 19) {
    float acc = 0.f;
    for (int c = 0; c < 256; ++c)
      acc += r_w[c] * ((j < 18) ? h_w[c * 18 + j] : h_b[c]);
    webe[j] = acc;  // [0..17] = effective 2x3x3 weights, [18] = effective bias
  }
}

__global__ __launch_bounds__(128) void vin_main_kernel(
    const float* __restrict__ X, const float* __restrict__ S1,
    const float* __restrict__ S2, const float* __restrict__ gamma,
    const float* __restrict__ q_w, const float* __restrict__ wkern,
    const float* __restrict__ fc1, const float* __restrict__ fc2,
    const float* __restrict__ fc3, const float* __restrict__ webe,
    float* __restrict__ out) {
  __shared__ float vpad[18 * 18 + 1];  // padded v (initially r); [324] == 0
  __shared__ float xpad[2 * 18 * 18];  // zero-padded X
  __shared__ float vout[579];          // gathered features + S1,S2,gamma
  __shared__ float l1s[100];
  __shared__ float l2s[64];

  const int b = blockIdx.x;
  const int tid = threadIdx.x;
  const int wave = tid >> 5;
  const int lane = tid & 31;
  const int hi = lane >> 4;  // lane half (K/M-half selector in fragments)
  const int n = lane & 15;   // pixel-in-row / M index
  const int bperm = ((lane ^ 16) << 2);  // ds_bpermute byte address

  // ---- init padded buffers -------------------------------------------------
  for (int i = tid; i < 18 * 18 + 1; i += 128) vpad[i] = 0.f;
  for (int i = tid; i < 2 * 18 * 18; i += 128) xpad[i] = 0.f;
  __syncthreads();
  const float* Xb = X + (size_t)b * 512;
  for (int i = tid; i < 512; i += 128) {
    int ch = i >> 8, p = i & 255;
    xpad[ch * 324 + ((p >> 4) + 1) * 18 + (p & 15) + 1] = Xb[i];
  }
  __syncthreads();

  // ---- r = conv3x3(X, folded_w) + folded_b  -> vpad interior ---------------
  float we[18];
#pragma unroll
  for (int j = 0; j < 18; ++j) we[j] = webe[j];
  const float be = webe[18];
  for (int p = tid; p < 256; p += 128) {
    int y = p >> 4, x = p & 15;
    float acc = be;
#pragma unroll
    for (int ci = 0; ci < 2; ++ci)
#pragma unroll
      for (int ky = 0; ky < 3; ++ky)
#pragma unroll
        for (int kx = 0; kx < 3; ++kx)
          acc += we[ci * 9 + ky * 3 + kx] *
                 xpad[ci * 324 + (y + ky) * 18 + (x + kx)];
    vpad[(y + 1) * 18 + x + 1] = acc;
  }
  __syncthreads();

  // ---- loop-invariant B-fragment LDS offsets (zero slot for K pad) --------
  // B-matrix 4x16 f32 layout (mirror of A): lane = (k>=2?16:0)+n ; vgpr = k&1.
  // P[k][pixel(y,n)] = v_pad[y+dy][n+dx], k -> (dy,dx) of the 3x3 stencil.
  int boff[4][6];
#pragma unroll
  for (int j = 0; j < 4; ++j) {
    int y = wave * 4 + j;
#pragma unroll
    for (int s = 0; s < 3; ++s)
#pragma unroll
      for (int r = 0; r < 2; ++r) {
        int k = 4 * s + 2 * hi + r;
        int dy = k / 3 - 1;
        int dx = k % 3 - 1;
        boff[j][2 * s + r] =
            (k < 9) ? ((y + dy + 1) * 18 + (n + dx + 1)) : (18 * 18);
      }
  }

  // ---- preload value-iteration kernel W as WMMA A-fragments ---------------
  v2f aw[4][3];
  load_afrag(wkern, hi, n, aw);

  v8f qr[4][4];  // [row j][mt] : qr C-fragments, live in VGPRs for all K iters
  float newv[4];

  // ---- pass 0: qr = conv3x3(r, q_w); v0 = max_ch(qr) ----------------------
  {
    v2f aq[4][3];
    load_afrag(q_w, hi, n, aq);
#pragma unroll
    for (int j = 0; j < 4; ++j) {
      v2f bf[3];
#pragma unroll
      for (int s = 0; s < 3; ++s) {
        bf[s][0] = vpad[boff[j][2 * s + 0]];
        bf[s][1] = vpad[boff[j][2 * s + 1]];
      }
      float rowmax = -3.402823466e38f;
#pragma unroll
      for (int mt = 0; mt < 4; ++mt) {
        v8f c = {};
        c = wmma4(aq[mt][0], bf[0], c);
        c = wmma4(aq[mt][1], bf[1], c);
        c = wmma4(aq[mt][2], bf[2], c);
        qr[j][mt] = c;
#pragma unroll
        for (int d = 0; d < 8; ++d) rowmax = fmaxf(rowmax, c[d]);
      }
      float om = __int_as_float(
          __builtin_amdgcn_ds_bpermute(bperm, __float_as_int(rowmax)));
      newv[j] = fmaxf(rowmax, om);
    }
  }
  __syncthreads();
  if (hi == 0)
#pragma unroll
    for (int j = 0; j < 4; ++j) vpad[(wave * 4 + j + 1) * 18 + n + 1] = newv[j];
  __syncthreads();

  // ---- K-1 = 39 middle iterations: q = qr + conv(v,w); v = max_ch(q) ------
  for (int it = 0; it < KITER - 1; ++it) {
#pragma unroll
    for (int j = 0; j < 4; ++j) {
      v2f bf[3];
#pragma unroll
      for (int s = 0; s < 3; ++s) {
        bf[s][0] = vpad[boff[j][2 * s + 0]];
        bf[s][1] = vpad[boff[j][2 * s + 1]];
      }
      float rowmax = -3.402823466e38f;
#pragma unroll
      for (int mt = 0; mt < 4; ++mt) {
        v8f c = qr[j][mt];
        c = wmma4(aw[mt][0], bf[0], c);
        c = wmma4(aw[mt][1], bf[1], c);
        c = wmma4(aw[mt][2], bf[2], c);
#pragma unroll
        for (int d = 0; d < 8; ++d) rowmax = fmaxf(rowmax, c[d]);
      }
      float om = __int_as_float(
          __builtin_amdgcn_ds_bpermute(bperm, __float_as_int(rowmax)));
      newv[j] = fmaxf(rowmax, om);
    }
    __syncthreads();
    if (hi == 0)
#pragma unroll
      for (int j = 0; j < 4; ++j)
        vpad[(wave * 4 + j + 1) * 18 + n + 1] = newv[j];
    __syncthreads();
  }

  // ---- final conv (#40) + 3x3 gather straight from D fragments ------------
  const float s1v = S1[b], s2v = S2[b], gv = gamma[b];
  int s1g = min(max((int)floorf(s1v), 0), 15);
  int s2g = min(max(15 - (int)floorf(s2v), 0), 15);
  int s1i[3] = {max(s1g - 1, 0), s1g, min(s1g + 1, 15)};
  int s2i[3] = {max(s2g - 1, 0), s2g, min(s2g + 1, 15)};
#pragma unroll
  for (int j = 0; j < 4; ++j) {
    int y = wave * 4 + j;
    v2f bf[3];
#pragma unroll
    for (int s = 0; s < 3; ++s) {
      bf[s][0] = vpad[boff[j][2 * s + 0]];
      bf[s][1] = vpad[boff[j][2 * s + 1]];
    }
    bool rowhit = (s2i[0] == y) | (s2i[1] == y) | (s2i[2] == y);
    bool colhit = (s1i[0] == n) | (s1i[1] == n) | (s1i[2] == n);
#pragma unroll
    for (int mt = 0; mt < 4; ++mt) {
      v8f c = qr[j][mt];
      c = wmma4(aw[mt][0], bf[0], c);
      c = wmma4(aw[mt][1], bf[1], c);
      c = wmma4(aw[mt][2], bf[2], c);
      if (rowhit && colhit) {
#pragma unroll
        for (int d = 0; d < 8; ++d) {
          int cch = 16 * mt + 8 * hi + d;
#pragma unroll
          for (int ii = 0; ii < 3; ++ii)
#pragma unroll
            for (int jj = 0; jj < 3; ++jj)
              if (s2i[jj] == y && s1i[ii] == n)
                vout[(ii * 3 + jj) * 64 + cch] = c[d];
        }
      }
    }
  }
  if (tid == 0) {
    vout[576] = s1v;
    vout[577] = s2v;
    vout[578] = gv;
  }
  __syncthreads();

  // ---- MLP: 579 -> 100 -> 50 -> 1 -----------------------------------------
  if (tid < 100) {
    float acc = 0.f;
    const float* wrow = fc1 + tid * 579;
    for (int k = 0; k < 579; ++k) acc += wrow[k] * vout[k];
    l1s[tid] = fmaxf(acc, 0.f);
  }
  __syncthreads();
  if (tid < 50) {
    float acc = 0.f;
    const float* wrow = fc2 + tid * 100;
#pragma unroll 4
    for (int k = 0; k < 100; ++k) acc += wrow[k] * l1s[k];
    l2s[tid] = fmaxf(acc, 0.f);
  }
  __syncthreads();
  if (tid == 0) {
    float acc = 0.f;
    for (int k = 0; k < 50; ++k) acc += fc3[k] * l2s[k];
    out[b] = acc;
  }
}

extern "C" void kernel_launch(void* const* d_in, const int* in_sizes, int n_in,
                              void* d_out, int out_size, void* d_ws,
                              size_t ws_size, hipStream_t stream) {
  const float* X = (const float*)d_in[0];
  const float* S1 = (const float*)d_in[1];
  const float* S2 = (const float*)d_in[2];
  const float* gamma = (const float*)d_in[3];
  const float* h_w = (const float*)d_in[4];
  const float* h_b = (const float*)d_in[5];
  const float* r_w = (const float*)d_in[6];
  const float* q_w = (const float*)d_in[7];
  const float* w = (const float*)d_in[8];
  const float* fc1 = (const float*)d_in[9];
  const float* fc2 = (const float*)d_in[10];
  const float* fc3 = (const float*)d_in[11];
  float* out = (float*)d_out;
  float* webe = (float*)d_ws;  // 19 floats of folded conv weights

  vin_prep_kernel<<<1, 32, 0, stream>>>(h_w, h_b, r_w, webe);
  vin_main_kernel<<<1024, 128, 0, stream>>>(X, S1, S2, gamma, q_w, w, fc1, fc2,
                                            fc3, webe, out);
  (void)in_sizes;
  (void)n_in;
  (void)out_size;
  (void)ws_size;
}